// AttnDecoderRNN2_34591666602137
// MI455X (gfx1250) — compile-verified
//
#include <hip/hip_runtime.h>
#include <math.h>

// Problem constants (match reference)
#define BB 64
#define SS 256
#define HH 512
#define VV 512

typedef float v2f __attribute__((ext_vector_type(2)));
typedef float v8f __attribute__((ext_vector_type(8)));

#define WMMA_F32(a, b, c) \
    __builtin_amdgcn_wmma_f32_16x16x4_f32(false, (a), false, (b), (short)0, (c), false, false)

// ---------------------------------------------------------------------------
// Skinny GEMM:  C[m,j] = sum_k A[m,k] * W[j,k] + bias[j]
// A: M x K row-major (M multiple of 64), W: N x K row-major (C = A*W^T + b).
// One wave computes a 64x16 strip = 4 row-tiles sharing one B fragment per
// K-step. Software-pipelined with prefetch distance 2 so WMMAs never wait on
// the loads of their own K-step (staggered s_wait_loadcnt instead of 0x0).
// ---------------------------------------------------------------------------
__global__ void gemm_awt_kernel(const float* __restrict__ A,
                                const float* __restrict__ W,
                                const float* __restrict__ bias,
                                float* __restrict__ C,
                                int M, int N, int K, int totalStrips) {
    const int wavesPerBlock = blockDim.x >> 5;
    const int wid   = threadIdx.x >> 5;
    const int strip = blockIdx.x * wavesPerBlock + wid;
    if (strip >= totalStrips) return;            // wave-uniform guard

    const int nTiles  = N >> 4;
    const int rowGrp  = strip / nTiles;          // group of 64 rows
    const int colTile = strip % nTiles;

    const int lane = threadIdx.x & 31;
    const int idx  = lane & 15;                  // M row (A) / N col (B)
    const int kh   = (lane >> 4) << 1;           // K sub-offset 0 or 2

    const float* Ap = A + (size_t)(rowGrp * 64 + idx) * K + kh;
    const float* Wp = W + (size_t)(colTile * 16 + idx) * K + kh;
    const size_t rs = (size_t)16 * K;            // next 16-row tile

    // Fragment sets for K-steps k, k+4 (prefetch distance 2)
    v2f b0  = *(const v2f*)(Wp);
    v2f a00 = *(const v2f*)(Ap);
    v2f a01 = *(const v2f*)(Ap + rs);
    v2f a02 = *(const v2f*)(Ap + 2 * rs);
    v2f a03 = *(const v2f*)(Ap + 3 * rs);
    v2f b1  = *(const v2f*)(Wp + 4);
    v2f a10 = *(const v2f*)(Ap + 4);
    v2f a11 = *(const v2f*)(Ap + rs + 4);
    v2f a12 = *(const v2f*)(Ap + 2 * rs + 4);
    v2f a13 = *(const v2f*)(Ap + 3 * rs + 4);

    v8f acc0 = {}, acc1 = {}, acc2 = {}, acc3 = {};
    int k = 0;
#pragma unroll 2
    for (; k < K - 8; k += 4) {
        // issue k+8 loads first; the k-set below is already resident
        v2f b2  = *(const v2f*)(Wp + k + 8);
        v2f a20 = *(const v2f*)(Ap + k + 8);
        v2f a21 = *(const v2f*)(Ap + rs + k + 8);
        v2f a22 = *(const v2f*)(Ap + 2 * rs + k + 8);
        v2f a23 = *(const v2f*)(Ap + 3 * rs + k + 8);
        acc0 = WMMA_F32(a00, b0, acc0);
        acc1 = WMMA_F32(a01, b0, acc1);
        acc2 = WMMA_F32(a02, b0, acc2);
        acc3 = WMMA_F32(a03, b0, acc3);
        b0 = b1;  a00 = a10; a01 = a11; a02 = a12; a03 = a13;
        b1 = b2;  a10 = a20; a11 = a21; a12 = a22; a13 = a23;
    }
    // drain the two in-flight sets
    acc0 = WMMA_F32(a00, b0, acc0);
    acc1 = WMMA_F32(a01, b0, acc1);
    acc2 = WMMA_F32(a02, b0, acc2);
    acc3 = WMMA_F32(a03, b0, acc3);
    acc0 = WMMA_F32(a10, b1, acc0);
    acc1 = WMMA_F32(a11, b1, acc1);
    acc2 = WMMA_F32(a12, b1, acc2);
    acc3 = WMMA_F32(a13, b1, acc3);

    // Store: VGPR r, lane l -> [M = r + 8*(l>=16)][N = l&15]
    const int n  = lane & 15;
    const int mb = (lane >> 4) << 3;
    const float bval = bias[colTile * 16 + n];
    float* Cp = C + (size_t)(rowGrp * 64) * N + colTile * 16;
#pragma unroll
    for (int r = 0; r < 8; ++r) {
        Cp[(size_t)(mb + r) * N + n]      = acc0[r] + bval;
        Cp[(size_t)(mb + r + 16) * N + n] = acc1[r] + bval;
        Cp[(size_t)(mb + r + 32) * N + n] = acc2[r] + bval;
        Cp[(size_t)(mb + r + 48) * N + n] = acc3[r] + bval;
    }
}

// ---------------------------------------------------------------------------
// Attention: one block per batch row b.
//   scores[s] = sum_h tanh(q[b,h] + kp[b,s,h]) * va[h] + bv
//   w = softmax(scores);  ctx[b,h] = sum_s w[s] * enc[b,s,h]
// VALU/transcendental bound; kp+enc stream stays resident in the 192MB L2.
// ---------------------------------------------------------------------------
__global__ void attn_kernel(const float* __restrict__ kp,
                            const float* __restrict__ enc,
                            const float* __restrict__ q,
                            const float* __restrict__ Va,
                            const float* __restrict__ bv,
                            float* __restrict__ ctx,
                            float* __restrict__ att_out,
                            int t) {
    __shared__ float qs[HH];
    __shared__ float vas[HH];
    __shared__ float red[SS];
    __shared__ float wsm[SS];

    const int b   = blockIdx.x;
    const int tid = threadIdx.x;           // 0..255, also the key index s

    qs[tid]        = q[b * HH + tid];
    qs[tid + 256]  = q[b * HH + tid + 256];
    vas[tid]       = Va[tid];
    vas[tid + 256] = Va[tid + 256];
    __syncthreads();

    // Warm L2/L0 for the context phase's strided enc reads (global_prefetch).
    __builtin_prefetch(enc + ((size_t)b * SS + tid) * HH, 0, 1);

    // energy score for s = tid (float4-vectorized over H)
    const float* kprow = kp + ((size_t)b * SS + tid) * HH;
    float acc = 0.f;
    for (int h = 0; h < HH; h += 4) {
        float4 kv = *(const float4*)(kprow + h);
        acc += tanhf(qs[h + 0] + kv.x) * vas[h + 0]
             + tanhf(qs[h + 1] + kv.y) * vas[h + 1]
             + tanhf(qs[h + 2] + kv.z) * vas[h + 2]
             + tanhf(qs[h + 3] + kv.w) * vas[h + 3];
    }
    acc += bv[0];

    // softmax over S=256 (LDS tree reductions)
    red[tid] = acc; __syncthreads();
    for (int off = 128; off > 0; off >>= 1) {
        if (tid < off) red[tid] = fmaxf(red[tid], red[tid + off]);
        __syncthreads();
    }
    const float mx = red[0]; __syncthreads();
    const float e = expf(acc - mx);
    red[tid] = e; __syncthreads();
    for (int off = 128; off > 0; off >>= 1) {
        if (tid < off) red[tid] += red[tid + off];
        __syncthreads();
    }
    const float w = e / red[0];
    wsm[tid] = w;
    att_out[((size_t)b * SS + t) * SS + tid] = w;   // attentions[b, t, s]
    __syncthreads();

    // ctx[b,h] = sum_s w[s] * enc[b,s,h]  (coalesced across threads per s)
    for (int h = tid; h < HH; h += 256) {
        float c = 0.f;
        const float* eb = enc + (size_t)b * SS * HH + h;
        for (int s = 0; s < SS; ++s) c += wsm[s] * eb[(size_t)s * HH];
        ctx[b * HH + h] = c;
    }
}

// ---------------------------------------------------------------------------
// Fused GRU gates: one wave per 32x16 strip (2 row-tiles) of the hidden state.
// Twelve WMMA accumulations share six weight fragments per K-step; software-
// pipelined with prefetch distance 1 (register budget: 12 accs + 2x10 frags).
// Gate nonlinearity applied in-registers, h_new written directly.
// ---------------------------------------------------------------------------
__global__ void gru_gate_kernel(const float* __restrict__ ctx,
                                const float* __restrict__ hprev,
                                const float* __restrict__ W_ih,
                                const float* __restrict__ b_ih,
                                const float* __restrict__ W_hh,
                                const float* __restrict__ b_hh,
                                float* __restrict__ hnext) {
    const int wid  = threadIdx.x >> 5;
    const int wave = blockIdx.x * (blockDim.x >> 5) + wid;   // 0..63
    const int rowGrp  = wave & 1;        // 2 groups of 32 rows
    const int colTile = wave >> 1;       // 512 hidden cols / 16

    const int lane = threadIdx.x & 31;
    const int idx  = lane & 15;
    const int kh   = (lane >> 4) << 1;
    const int j0   = colTile * 16;

    const float* ac  = ctx   + (size_t)(rowGrp * 32 + idx) * HH + kh;
    const float* ah  = hprev + (size_t)(rowGrp * 32 + idx) * HH + kh;
    const float* wir = W_ih + (size_t)(j0 + idx) * HH + kh;
    const float* wiz = wir + (size_t)512 * HH;
    const float* win = wir + (size_t)1024 * HH;
    const float* whr = W_hh + (size_t)(j0 + idx) * HH + kh;
    const float* whz = whr + (size_t)512 * HH;
    const float* whn = whr + (size_t)1024 * HH;
    const size_t rs = (size_t)16 * HH;

    // current fragment set (10 fragments)
    v2f cir = *(const v2f*)(wir);
    v2f ciz = *(const v2f*)(wiz);
    v2f cin = *(const v2f*)(win);
    v2f chr = *(const v2f*)(whr);
    v2f chz = *(const v2f*)(whz);
    v2f chn = *(const v2f*)(whn);
    v2f cc0 = *(const v2f*)(ac);
    v2f cc1 = *(const v2f*)(ac + rs);
    v2f ch0 = *(const v2f*)(ah);
    v2f ch1 = *(const v2f*)(ah + rs);

    v8f air[2] = {{}, {}}, aiz[2] = {{}, {}}, ain[2] = {{}, {}};
    v8f ahr[2] = {{}, {}}, ahz[2] = {{}, {}}, ahn[2] = {{}, {}};

    for (int k = 0; k < HH - 4; k += 4) {
        // prefetch next K-step's fragments before consuming current ones
        v2f nir = *(const v2f*)(wir + k + 4);
        v2f niz = *(const v2f*)(wiz + k + 4);
        v2f nin = *(const v2f*)(win + k + 4);
        v2f nhr = *(const v2f*)(whr + k + 4);
        v2f nhz = *(const v2f*)(whz + k + 4);
        v2f nhn = *(const v2f*)(whn + k + 4);
        v2f nc0 = *(const v2f*)(ac + k + 4);
        v2f nc1 = *(const v2f*)(ac + rs + k + 4);
        v2f nh0 = *(const v2f*)(ah + k + 4);
        v2f nh1 = *(const v2f*)(ah + rs + k + 4);

        air[0] = WMMA_F32(cc0, cir, air[0]);  air[1] = WMMA_F32(cc1, cir, air[1]);
        aiz[0] = WMMA_F32(cc0, ciz, aiz[0]);  aiz[1] = WMMA_F32(cc1, ciz, aiz[1]);
        ain[0] = WMMA_F32(cc0, cin, ain[0]);  ain[1] = WMMA_F32(cc1, cin, ain[1]);
        ahr[0] = WMMA_F32(ch0, chr, ahr[0]);  ahr[1] = WMMA_F32(ch1, chr, ahr[1]);
        ahz[0] = WMMA_F32(ch0, chz, ahz[0]);  ahz[1] = WMMA_F32(ch1, chz, ahz[1]);
        ahn[0] = WMMA_F32(ch0, chn, ahn[0]);  ahn[1] = WMMA_F32(ch1, chn, ahn[1]);

        cir = nir; ciz = niz; cin = nin; chr = nhr; chz = nhz; chn = nhn;
        cc0 = nc0; cc1 = nc1; ch0 = nh0; ch1 = nh1;
    }
    // drain last K-step
    air[0] = WMMA_F32(cc0, cir, air[0]);  air[1] = WMMA_F32(cc1, cir, air[1]);
    aiz[0] = WMMA_F32(cc0, ciz, aiz[0]);  aiz[1] = WMMA_F32(cc1, ciz, aiz[1]);
    ain[0] = WMMA_F32(cc0, cin, ain[0]);  ain[1] = WMMA_F32(cc1, cin, ain[1]);
    ahr[0] = WMMA_F32(ch0, chr, ahr[0]);  ahr[1] = WMMA_F32(ch1, chr, ahr[1]);
    ahz[0] = WMMA_F32(ch0, chz, ahz[0]);  ahz[1] = WMMA_F32(ch1, chz, ahz[1]);
    ahn[0] = WMMA_F32(ch0, chn, ahn[0]);  ahn[1] = WMMA_F32(ch1, chn, ahn[1]);

    const int n  = lane & 15;
    const int mb = (lane >> 4) << 3;
    const int j  = j0 + n;
    const float bir = b_ih[j], biz = b_ih[j + 512], bin = b_ih[j + 1024];
    const float bhr = b_hh[j], bhz = b_hh[j + 512], bhn = b_hh[j + 1024];
#pragma unroll
    for (int rt = 0; rt < 2; ++rt) {
#pragma unroll
        for (int r = 0; r < 8; ++r) {
            const int m = rowGrp * 32 + rt * 16 + mb + r;
            const float rg = 1.f / (1.f + expf(-(air[rt][r] + bir + ahr[rt][r] + bhr)));
            const float zg = 1.f / (1.f + expf(-(aiz[rt][r] + biz + ahz[rt][r] + bhz)));
            const float ng = tanhf(ain[rt][r] + bin + rg * (ahn[rt][r] + bhn));
            const float hp = hprev[(size_t)m * HH + j];
            hnext[(size_t)m * HH + j] = (1.f - zg) * ng + zg * hp;
        }
    }
}

// ---------------------------------------------------------------------------
// In-place log_softmax over rows of V=512; one block per (t,b) row.
// ---------------------------------------------------------------------------
__global__ void logsoftmax_kernel(float* __restrict__ logits) {
    __shared__ float red[256];
    float* p = logits + (size_t)blockIdx.x * VV;
    const int tid = threadIdx.x;
    const float a = p[tid], b = p[tid + 256];

    red[tid] = fmaxf(a, b); __syncthreads();
    for (int off = 128; off > 0; off >>= 1) {
        if (tid < off) red[tid] = fmaxf(red[tid], red[tid + off]);
        __syncthreads();
    }
    const float mx = red[0]; __syncthreads();
    red[tid] = expf(a - mx) + expf(b - mx); __syncthreads();
    for (int off = 128; off > 0; off >>= 1) {
        if (tid < off) red[tid] += red[tid + off];
        __syncthreads();
    }
    const float l = logf(red[0]);
    p[tid]       = a - mx - l;
    p[tid + 256] = b - mx - l;
}

__global__ void zero_kernel(float* __restrict__ p, int n) {
    const int i = blockIdx.x * blockDim.x + threadIdx.x;
    if (i < n) p[i] = 0.f;
}

__global__ void copy_kernel(const float* __restrict__ src, float* __restrict__ dst, int n) {
    const int i = blockIdx.x * blockDim.x + threadIdx.x;
    if (i < n) dst[i] = src[i];
}

// ---------------------------------------------------------------------------
extern "C" void kernel_launch(void* const* d_in, const int* in_sizes, int n_in,
                              void* d_out, int out_size, void* d_ws, size_t ws_size,
                              hipStream_t stream) {
    (void)in_sizes; (void)n_in; (void)out_size; (void)ws_size;

    const float* enc  = (const float*)d_in[0];
    const float* Wa   = (const float*)d_in[1];
    const float* ba   = (const float*)d_in[2];
    const float* Ua   = (const float*)d_in[3];
    const float* bu   = (const float*)d_in[4];
    const float* Va   = (const float*)d_in[5];
    const float* bv   = (const float*)d_in[6];
    const float* W_ih = (const float*)d_in[7];
    const float* b_ih = (const float*)d_in[8];
    const float* W_hh = (const float*)d_in[9];
    const float* b_hh = (const float*)d_in[10];
    const float* Wout = (const float*)d_in[11];
    const float* bout = (const float*)d_in[12];

    // Workspace layout (floats)
    float* ws  = (float*)d_ws;
    float* kp  = ws;                                   // [B,S,H] keys_proj (32 MB)
    float* h0  = kp + (size_t)BB * SS * HH;            // [B,H] ping
    float* h1  = h0 + (size_t)BB * HH;                 // [B,H] pong
    float* q   = h1 + (size_t)BB * HH;                 // [B,H]
    float* ctx = q  + (size_t)BB * HH;                 // [B,H]

    // Output layout: log_probs [S,B,V] | h_final [1,B,H] | attentions [B,S,S]
    float* out  = (float*)d_out;
    float* logp = out;
    float* hfin = out + (size_t)SS * BB * VV;
    float* att  = hfin + (size_t)BB * HH;

    // h0 = 0
    zero_kernel<<<(BB * HH + 255) / 256, 256, 0, stream>>>(h0, BB * HH);

    // keys_proj = enc @ Ua^T + bu : (16384/64)*(512/16) = 8192 strips
    gemm_awt_kernel<<<1024, 256, 0, stream>>>(enc, Ua, bu, kp,
                                              BB * SS, HH, HH, 8192);

    for (int t = 0; t < SS; ++t) {
        float* hin  = (t & 1) ? h1 : h0;
        float* hout = (t & 1) ? h0 : h1;

        // q = h @ Wa^T + ba : (64/64)*(512/16) = 32 strips
        gemm_awt_kernel<<<4, 256, 0, stream>>>(hin, Wa, ba, q, BB, HH, HH, 32);

        // attention + softmax + context
        attn_kernel<<<BB, 256, 0, stream>>>(kp, enc, q, Va, bv, ctx, att, t);

        // fused GRU gates -> h_new : 64 waves
        gru_gate_kernel<<<8, 256, 0, stream>>>(ctx, hin, W_ih, b_ih,
                                               W_hh, b_hh, hout);

        // logits = h_new @ Wout^T + bout, written straight into d_out[t]
        gemm_awt_kernel<<<4, 256, 0, stream>>>(hout, Wout, bout,
                                               logp + (size_t)t * BB * VV,
                                               BB, VV, HH, 32);
    }

    // log_softmax over all S*B rows of V
    logsoftmax_kernel<<<SS * BB, 256, 0, stream>>>(logp);

    // final hidden state: after 256 steps the last write landed in h0
    copy_kernel<<<(BB * HH + 255) / 256, 256, 0, stream>>>(h0, hfin, BB * HH);
}